// SparseAttention_36601711296570
// MI455X (gfx1250) — compile-verified
//
#include <hip/hip_runtime.h>
#include <hip/hip_bf16.h>

typedef float v2f __attribute__((ext_vector_type(2)));
typedef float v8f __attribute__((ext_vector_type(8)));

#define D 256
#define H 8

// ---------------- monotone float <-> uint for atomic max ----------------
__device__ __forceinline__ unsigned enc_f32(float f) {
    unsigned u = __float_as_uint(f);
    return (u & 0x80000000u) ? ~u : (u | 0x80000000u);
}
__device__ __forceinline__ float dec_f32(unsigned u) {
    unsigned v = (u & 0x80000000u) ? (u & 0x7fffffffu) : ~u;
    return __uint_as_float(v);
}

// ---------------- Fused QKV projection (WMMA f32).
// One wave per 16x16 output tile; each A-fragment is reused for 3 WMMAs.
__global__ __launch_bounds__(256) void gat_qkv_wmma(
    const float* __restrict__ A,
    const float* __restrict__ Wq, const float* __restrict__ biasq,
    const float* __restrict__ Wk, const float* __restrict__ biask,
    const float* __restrict__ Wv, const float* __restrict__ biasv,
    float* __restrict__ Q, float* __restrict__ Kf, float* __restrict__ Vf,
    int M, float qscale)
{
    const int lane = threadIdx.x & 31;
    const int wave = threadIdx.x >> 5;
    const int tile = blockIdx.x * 8 + wave;
    const int tiles_n = D / 16;                 // 16
    const int tm = tile / tiles_n;
    const int tn = tile % tiles_n;
    if (tm * 16 >= M) return;                   // wave-uniform exit

    const int half = lane >> 4;                 // 0: K pair {0,1}, 1: {2,3}
    const int l16  = lane & 15;
    int arow = tm * 16 + l16;
    if (arow > M - 1) arow = M - 1;             // clamp, keep EXEC all-ones
    const int bcol = tn * 16 + l16;

    const float* __restrict__ arowp = A + (size_t)arow * D;
    v8f cq = {}, ck = {}, cv = {};
#pragma unroll 4
    for (int k = 0; k < D; k += 4) {
        const int ka = k + 2 * half;
        v2f a;  a.x  = arowp[ka];          a.y  = arowp[ka + 1];
        v2f bq; bq.x = Wq[(size_t)ka * D + bcol]; bq.y = Wq[(size_t)(ka + 1) * D + bcol];
        v2f bk; bk.x = Wk[(size_t)ka * D + bcol]; bk.y = Wk[(size_t)(ka + 1) * D + bcol];
        v2f bv; bv.x = Wv[(size_t)ka * D + bcol]; bv.y = Wv[(size_t)(ka + 1) * D + bcol];
        cq = __builtin_amdgcn_wmma_f32_16x16x4_f32(false, a, false, bq, (short)0, cq, false, false);
        ck = __builtin_amdgcn_wmma_f32_16x16x4_f32(false, a, false, bk, (short)0, ck, false, false);
        cv = __builtin_amdgcn_wmma_f32_16x16x4_f32(false, a, false, bv, (short)0, cv, false, false);
    }
    const float vq = biasq[bcol], vk = biask[bcol], vv = biasv[bcol];
    const size_t coloff = (size_t)bcol;
    if (tm * 16 + 16 <= M) {                    // wave-uniform: full tile (always for N%16==0)
#pragma unroll
        for (int j = 0; j < 8; ++j) {
            const size_t m = (size_t)(tm * 16 + j + 8 * half);
            Q [m * D + coloff] = (cq[j] + vq) * qscale;
            Kf[m * D + coloff] = (ck[j] + vk);
            Vf[m * D + coloff] = (cv[j] + vv);
        }
    } else {
#pragma unroll
        for (int j = 0; j < 8; ++j) {
            const int m = tm * 16 + j + 8 * half;
            if (m < M) {
                Q [(size_t)m * D + coloff] = (cq[j] + vq) * qscale;
                Kf[(size_t)m * D + coloff] = (ck[j] + vk);
                Vf[(size_t)m * D + coloff] = (cv[j] + vv);
            }
        }
    }
}

// ---------------- Output projection (WMMA f32): Out = A @ W + b
__global__ __launch_bounds__(256) void gat_gemm_wmma(
    const float* __restrict__ A, const float* __restrict__ W,
    const float* __restrict__ bias, float* __restrict__ Out, int M)
{
    const int lane = threadIdx.x & 31;
    const int wave = threadIdx.x >> 5;
    const int tile = blockIdx.x * 8 + wave;
    const int tiles_n = D / 16;
    const int tm = tile / tiles_n;
    const int tn = tile % tiles_n;
    if (tm * 16 >= M) return;

    const int half = lane >> 4;
    const int l16  = lane & 15;
    int arow = tm * 16 + l16;
    if (arow > M - 1) arow = M - 1;
    const int bcol = tn * 16 + l16;

    const float* __restrict__ arowp = A + (size_t)arow * D;
    v8f c = {};
#pragma unroll 4
    for (int k = 0; k < D; k += 4) {
        const int ka = k + 2 * half;
        v2f a; a.x = arowp[ka];               a.y = arowp[ka + 1];
        v2f b; b.x = W[(size_t)ka * D + bcol]; b.y = W[(size_t)(ka + 1) * D + bcol];
        c = __builtin_amdgcn_wmma_f32_16x16x4_f32(false, a, false, b, (short)0, c, false, false);
    }
    const float bvv = bias[bcol];
    if (tm * 16 + 16 <= M) {
#pragma unroll
        for (int j = 0; j < 8; ++j) {
            const size_t m = (size_t)(tm * 16 + j + 8 * half);
            Out[m * D + bcol] = c[j] + bvv;
        }
    } else {
#pragma unroll
        for (int j = 0; j < 8; ++j) {
            const int m = tm * 16 + j + 8 * half;
            if (m < M) Out[(size_t)m * D + bcol] = c[j] + bvv;
        }
    }
}

// ---------------- SDDMM + running segment max.  One wave per edge.
// Lane owns 8 contiguous features = one value per head (f % 8 == h).
__global__ __launch_bounds__(256) void gat_sddmm_max(
    const float* __restrict__ Q, const float* __restrict__ Kf,
    const int* __restrict__ row, const int* __restrict__ col,
    float* __restrict__ S, unsigned* __restrict__ Mx, int E)
{
    const int lane = threadIdx.x & 31;
    const int wid  = blockIdx.x * (blockDim.x >> 5) + (threadIdx.x >> 5);
    const int nw   = gridDim.x * (blockDim.x >> 5);
    for (int e = wid; e < E; e += nw) {
        const int r = row[e], cc = col[e];
        const float4* q4 = (const float4*)(Q  + (size_t)r  * D);
        const float4* k4 = (const float4*)(Kf + (size_t)cc * D);
        const float4 qa = q4[2 * lane], qb = q4[2 * lane + 1];
        const float4 ka = k4[2 * lane], kb = k4[2 * lane + 1];
        float acc[8] = { qa.x * ka.x, qa.y * ka.y, qa.z * ka.z, qa.w * ka.w,
                         qb.x * kb.x, qb.y * kb.y, qb.z * kb.z, qb.w * kb.w };
#pragma unroll
        for (int j = 0; j < 8; ++j)
#pragma unroll
            for (int off = 16; off >= 1; off >>= 1)
                acc[j] += __shfl_xor(acc[j], off, 32);
        if (lane == 0) {
#pragma unroll
            for (int j = 0; j < 8; ++j) {
                S[(size_t)e * H + j] = acc[j];
                atomicMax(&Mx[(size_t)r * H + j], enc_f32(acc[j]));
            }
        }
    }
}

// ---------------- exp(s - m) in place + segment sum Z.  One thread per (e,h).
__global__ __launch_bounds__(256) void gat_exp_z(
    const int* __restrict__ row, float* __restrict__ S,
    const unsigned* __restrict__ Mx, float* __restrict__ Z, long long EH)
{
    const long long t = (long long)blockIdx.x * blockDim.x + threadIdx.x;
    if (t >= EH) return;
    const int e = (int)(t >> 3);
    const int h = (int)(t & 7);
    const int r = row[e];
    const float m = dec_f32(Mx[(size_t)r * H + h]);
    const float v = __expf(S[t] - m);
    S[t] = v;
    atomicAdd(&Z[(size_t)r * H + h], v);
}

// ---------------- SPMM scatter:  Acc[row[e], f] += e_val[e, f%8] * V[col[e], f]
__global__ __launch_bounds__(256) void gat_spmm(
    const float* __restrict__ Vf,
    const int* __restrict__ row, const int* __restrict__ col,
    const float* __restrict__ S, float* __restrict__ Acc, int E)
{
    const int lane = threadIdx.x & 31;
    const int wid  = blockIdx.x * (blockDim.x >> 5) + (threadIdx.x >> 5);
    const int nw   = gridDim.x * (blockDim.x >> 5);
    for (int e = wid; e < E; e += nw) {
        const int r = row[e], cc = col[e];
        const float4* v4 = (const float4*)(Vf + (size_t)cc * D);
        const float4 va = v4[2 * lane], vb = v4[2 * lane + 1];
        const float4* s4 = (const float4*)(S + (size_t)e * H);
        const float4 sa = s4[0], sb = s4[1];
        float* dst = Acc + (size_t)r * D + lane * 8;
        atomicAdd(dst + 0, sa.x * va.x);
        atomicAdd(dst + 1, sa.y * va.y);
        atomicAdd(dst + 2, sa.z * va.z);
        atomicAdd(dst + 3, sa.w * va.w);
        atomicAdd(dst + 4, sb.x * vb.x);
        atomicAdd(dst + 5, sb.y * vb.y);
        atomicAdd(dst + 6, sb.z * vb.z);
        atomicAdd(dst + 7, sb.w * vb.w);
    }
}

// ---------------- normalize:  Hb[i,f] = Acc[i,f] / Z[i, f%8]
__global__ __launch_bounds__(256) void gat_norm(
    const float* __restrict__ Acc, const float* __restrict__ Z,
    float* __restrict__ Hb, long long ND)
{
    const long long t = (long long)blockIdx.x * blockDim.x + threadIdx.x;
    if (t >= ND) return;
    const int i = (int)(t >> 8);          // /256
    const int h = (int)(t & 7);           // f % 8
    const float z = Z[(size_t)i * H + h];
    Hb[t] = (z > 0.0f) ? (Acc[t] / z) : 0.0f;
}

extern "C" void kernel_launch(void* const* d_in, const int* in_sizes, int n_in,
                              void* d_out, int out_size, void* d_ws, size_t ws_size,
                              hipStream_t stream) {
    const float* x   = (const float*)d_in[0];
    const int*   row = (const int*)  d_in[1];
    const int*   col = (const int*)  d_in[2];
    const float* Wq  = (const float*)d_in[3];
    const float* bq  = (const float*)d_in[4];
    const float* Wk  = (const float*)d_in[5];
    const float* bk  = (const float*)d_in[6];
    const float* Wv  = (const float*)d_in[7];
    const float* bv  = (const float*)d_in[8];
    const float* Wo  = (const float*)d_in[9];
    const float* bo  = (const float*)d_in[10];
    float* out = (float*)d_out;

    const int N = in_sizes[0] / D;
    const int E = in_sizes[1];
    const size_t ND = (size_t)N * D;

    // Workspace layout (ACC aliases Q, Hb aliases Kf — dead after SDDMM):
    float*    Q   = (float*)d_ws;
    float*    Kf  = Q + ND;
    float*    Vf  = Kf + ND;
    float*    S   = Vf + ND;                      // E*H scores -> exp values
    unsigned* Mx  = (unsigned*)(S + (size_t)E * H);
    float*    Z   = (float*)(Mx + (size_t)N * H);
    float*    Acc = Q;
    float*    Hb  = Kf;

    const float scaling = 0.17677669529663688f;   // 32^-0.5

    const int tilesM = (N + 15) / 16;
    const int gemmBlocks = (tilesM * (D / 16) + 7) / 8;   // 8 waves (tiles) / block

    // Fused Q/K/V projection: 1 A-load feeds 3 WMMAs
    gat_qkv_wmma<<<gemmBlocks, 256, 0, stream>>>(
        x, Wq, bq, Wk, bk, Wv, bv, Q, Kf, Vf, N, scaling);

    hipMemsetAsync(Mx, 0, (size_t)N * H * sizeof(unsigned), stream);  // 0 < enc(-inf)
    hipMemsetAsync(Z,  0, (size_t)N * H * sizeof(float), stream);

    // per-edge scores + segment max (wave per edge)
    gat_sddmm_max<<<(E + 7) / 8, 256, 0, stream>>>(Q, Kf, row, col, S, Mx, E);

    hipMemsetAsync(Acc, 0, ND * sizeof(float), stream);   // Q is dead now

    const long long EH = (long long)E * H;
    gat_exp_z<<<(int)((EH + 255) / 256), 256, 0, stream>>>(row, S, Mx, Z, EH);

    // scatter a[e,h] * V[col[e]] into Acc (L2-resident atomics)
    gat_spmm<<<(E + 7) / 8, 256, 0, stream>>>(Vf, row, col, S, Acc, E);

    gat_norm<<<(int)((ND + 255) / 256), 256, 0, stream>>>(Acc, Z, Hb, (long long)ND);

    // output projection (WMMA f32) straight into d_out
    gat_gemm_wmma<<<gemmBlocks, 256, 0, stream>>>(Hb, Wo, bo, out, N);
    (void)n_in; (void)out_size; (void)ws_size;
}